// KroneckerLinear_58248346469065
// MI455X (gfx1250) — compile-verified
//
#include <hip/hip_runtime.h>

// KroneckerLinear on MI455X (gfx1250), compile-only tuned.
//
// out[n, a*64+b] = sum_{c,d} x[n, c*64+d] * f1[a,c] * f2[b,d] + bias[a*64+b]
//
// Factored (never build the 8192x8192 kron matrix):
//   Stage A: S[la, d]          = sum_c f1[a_base+la, c] * x[n, c*64+d]   (K=128)
//   Stage B: out[n,(a)*64 + b] = bias[a*64+b] + sum_d S[la, d]*f2[b, d]  (K=64)
// S (32x64 fp32 per block) lives in LDS -> HBM traffic ~ x(2MB)+out(2MB).
// Bandwidth-bound (~0.18us @ 23.3TB/s); 201 MFLOP fp32 via v_wmma_f32_16x16x4_f32.
//
// Grid: 256 blocks = 64 n  x  4 a-slices of 32 rows. 256 threads = 8 waves.
// Each wave computes one 16x16 WMMA tile per stage; zero divergence so EXEC
// stays all-ones (WMMA requirement).

typedef float v2f __attribute__((ext_vector_type(2)));
typedef float v8f __attribute__((ext_vector_type(8)));

__global__ __launch_bounds__(256) void kron_fused_kernel(
    const float* __restrict__ x,     // [64, 8192]
    const float* __restrict__ f1,    // [128, 128]
    const float* __restrict__ f2,    // [64, 64]
    const float* __restrict__ bias,  // [8192]
    float* __restrict__ out)         // [64, 8192]
{
  __shared__ float s_lds[32 * 64];   // S slice: 32 a-rows x 64 d-cols (8 KB)

  const int n      = blockIdx.x >> 2;         // batch index
  const int a_base = (blockIdx.x & 3) << 5;   // 32-row a-slice
  const int waveId = threadIdx.x >> 5;        // 0..7
  const int lane   = threadIdx.x & 31;
  const int colL   = lane & 15;               // column/row within 16-group
  const int hi     = lane >> 4;               // 0 = lanes 0..15, 1 = 16..31
  const int kOff   = hi * 2;                  // K-pair offset inside K=4 step

  // Per-wave 16x16 tile coordinates (same mapping both stages).
  const int la0 = (waveId >> 2) << 4;         // 0 or 16  (local a row)
  const int t0  = (waveId & 3) << 4;          // 0,16,32,48 (d col / b col)

  const float* xn = x + n * 8192;

  // ---------------- Stage A: S = f1_slice @ x_n  (M=32,N=64,K=128) --------
  {
    // A-frag: f1[a_base+la0+colL, k + kOff + {0,1}]  -> contiguous b64 loads
    const float* aPtr = f1 + (a_base + la0 + colL) * 128 + kOff;
    // B-frag: x[n, (k + kOff + {0,1})*64 + t0 + colL] -> stride-64 b32 loads
    const float* bPtr = xn + kOff * 64 + t0 + colL;

    v8f acc = {};
#pragma unroll
    for (int k = 0; k < 128; k += 4) {
      v2f afrag = *(const v2f*)(aPtr + k);
      v2f bfrag;
      bfrag.x = bPtr[k * 64];
      bfrag.y = bPtr[k * 64 + 64];
      acc = __builtin_amdgcn_wmma_f32_16x16x4_f32(
          /*neg_a=*/false, afrag, /*neg_b=*/false, bfrag,
          /*c_mod=*/(short)0, acc, /*reuse_a=*/false, /*reuse_b=*/false);
    }

    // D layout: VGPR r holds M = la0 + r + hi*8, N = t0 + colL
    float* sp = &s_lds[(la0 + hi * 8) * 64 + t0 + colL];
#pragma unroll
    for (int r = 0; r < 8; ++r)
      sp[r * 64] = acc[r];
  }

  __syncthreads();

  // ---------------- Stage B: out_slice = S @ f2^T + bias (M=32,N=64,K=64) --
  {
    // A-frag: S[la0+colL, k + kOff + {0,1}] from LDS -> ds_load_b64
    const float* aPtr = &s_lds[(la0 + colL) * 64 + kOff];
    // B-frag: B[k, b] = f2[b, k] -> f2[(t0+colL)*64 + k + kOff + {0,1}], b64
    const float* bPtr = f2 + (t0 + colL) * 64 + kOff;
    // C init straight from bias: row a = a_base + la0 + r + hi*8, col t0+colL
    const float* cPtr = bias + (a_base + la0 + hi * 8) * 64 + t0 + colL;

    v8f acc;
#pragma unroll
    for (int r = 0; r < 8; ++r)
      acc[r] = cPtr[r * 64];

#pragma unroll
    for (int k = 0; k < 64; k += 4) {
      v2f afrag = *(const v2f*)(aPtr + k);
      v2f bfrag = *(const v2f*)(bPtr + k);
      acc = __builtin_amdgcn_wmma_f32_16x16x4_f32(
          false, afrag, false, bfrag, (short)0, acc, false, false);
    }

    // out flat index: n*8192 + a*64 + b  ==  ((n*128 + a)*64 + b)
    float* op = out + n * 8192 + (a_base + la0 + hi * 8) * 64 + t0 + colL;
#pragma unroll
    for (int r = 0; r < 8; ++r)
      op[r * 64] = acc[r];
  }
}

extern "C" void kernel_launch(void* const* d_in, const int* in_sizes, int n_in,
                              void* d_out, int out_size, void* d_ws, size_t ws_size,
                              hipStream_t stream) {
  (void)in_sizes; (void)n_in; (void)out_size; (void)d_ws; (void)ws_size;
  const float* x    = (const float*)d_in[0];
  const float* f1   = (const float*)d_in[1];
  const float* f2   = (const float*)d_in[2];
  const float* bias = (const float*)d_in[3];
  float*       out  = (float*)d_out;

  // 256 blocks (64 n x 4 a-slices), 8 waves each; fully deterministic.
  kron_fused_kernel<<<dim3(256), dim3(256), 0, stream>>>(x, f1, f2, bias, out);
}